// DSTGAT_50680614093549
// MI455X (gfx1250) — compile-verified
//
#include <hip/hip_runtime.h>
#include <hip/hip_bf16.h>
#include <math.h>

typedef float v2f __attribute__((ext_vector_type(2)));
typedef float v8f __attribute__((ext_vector_type(8)));

// ---------- helpers ----------
__device__ __forceinline__ unsigned fenc(float f) {
  unsigned u = __float_as_uint(f);
  return (u & 0x80000000u) ? ~u : (u | 0x80000000u);   // order-preserving map for atomicMax
}
__device__ __forceinline__ float fdec(unsigned u) {
  return (u & 0x80000000u) ? __uint_as_float(u & 0x7FFFFFFFu) : __uint_as_float(~u);
}

__global__ void zero_u32(unsigned* p, int n) {
  int i = blockIdx.x * blockDim.x + threadIdx.x;
  if (i < n) p[i] = 0u;
}

// ---------- generic strided GEMM on V_WMMA_F32_16X16X4_F32 ----------
// C[M,Nc] = act(A[M,K] * B[K,Nc] + bias)
// AMODE 0: A[r,k] = A[r*lda + k]                       (row major)
// AMODE 1: A[r,k] = A[(r>>2)*lda + k*4 + (r&3)]        (NCT tensor, rows = (n,t))
// AMODE 2: A[r,k] = A[r*lda + k*4 + 3]                 (NCT tensor, last timestep)
// B[k,j]  = W[k*ldbk + j*ldbj]
// CMODE 0: C[r*ldc + j]   CMODE 1: C[(r>>2)*ldc + j*4 + (r&3)]
template<int AMODE, int CMODE, int ACT>
__global__ void gemm_wmma_f32(const float* __restrict__ A, const float* __restrict__ W,
                              const float* __restrict__ bias, float* __restrict__ C,
                              int M, int Nc, int K, int lda, int ldbk, int ldbj, int ldc)
{
  const int lane = threadIdx.x & 31;
  const int wave = (blockIdx.x * blockDim.x + threadIdx.x) >> 5;
  const int ntile = Nc >> 4;
  const int mi = wave / ntile;
  const int ni = wave - mi * ntile;
  if (mi * 16 >= M) return;                       // wave-uniform exit, EXEC stays full

  const int half = lane >> 4;                     // 0: lanes 0-15, 1: lanes 16-31
  const int l16  = lane & 15;

  int arow = mi * 16 + l16;  if (arow > M - 1)  arow = M - 1;    // clamp (no predication)
  int bcol = ni * 16 + l16;  if (bcol > Nc - 1) bcol = Nc - 1;

  v8f acc = {};
  for (int k0 = 0; k0 < K; k0 += 4) {
    const int ka = k0 + half * 2;                 // A frag: V0=K(0|2), V1=K(1|3) per half
    v2f a, b;
    if (AMODE == 0) {
      a.x = A[arow * lda + ka];
      a.y = A[arow * lda + ka + 1];
    } else if (AMODE == 1) {
      const float* ap = A + (arow >> 2) * lda + (arow & 3);
      a.x = ap[ka * 4];
      a.y = ap[ka * 4 + 4];
    } else {
      const float* ap = A + arow * lda + 3;
      a.x = ap[ka * 4];
      a.y = ap[ka * 4 + 4];
    }
    b.x = W[ka * ldbk + bcol * ldbj];             // B frag mirrors A layout
    b.y = W[(ka + 1) * ldbk + bcol * ldbj];
    acc = __builtin_amdgcn_wmma_f32_16x16x4_f32(false, a, false, b, (short)0, acc,
                                                false, false);
  }

  const int col = ni * 16 + l16;
  const float bv = (bias != nullptr && col < Nc) ? bias[col] : 0.0f;
  #pragma unroll
  for (int v = 0; v < 8; ++v) {
    const int row = mi * 16 + half * 8 + v;       // C layout: lane=col, VGPR v = row (+8 for hi half)
    if (row < M && col < Nc) {
      float y = acc[v] + bv;
      if (ACT == 1) y = fmaxf(y, 0.0f);
      if (CMODE == 0) C[row * ldc + col] = y;
      else            C[(row >> 2) * ldc + col * 4 + (row & 3)] = y;
    }
  }
}

// ---------- gated dilated conv: tanh(conv(x,w1)) * sigmoid(conv(x,w2)) ----------
__global__ void dconv_gate(const float* __restrict__ x,
                           const float* __restrict__ w1, const float* __restrict__ b1,
                           const float* __restrict__ w2, const float* __restrict__ b2,
                           float* __restrict__ out, int N, int cin, int cout, int dil)
{
  int idx = blockIdx.x * blockDim.x + threadIdx.x;      // over N*cout*4
  if (idx >= N * cout * 4) return;
  const int t = idx & 3;
  const int o = (idx >> 2) % cout;
  const int n = idx / (cout * 4);
  float s1 = b1[o], s2 = b2[o];
  const float* xn = x + (size_t)n * cin * 4;
  for (int c = 0; c < cin; ++c) {
    const float* wr1 = w1 + (o * cin + c) * 3;
    const float* wr2 = w2 + (o * cin + c) * 3;
    #pragma unroll
    for (int kk = 0; kk < 3; ++kk) {
      const int tt = t + (kk - 1) * dil;                // SAME padding, rhs_dilation
      if (tt >= 0 && tt < 4) {
        const float xv = xn[c * 4 + tt];
        s1 += xv * wr1[kk];
        s2 += xv * wr2[kk];
      }
    }
  }
  out[((size_t)n * cout + o) * 4 + t] = tanhf(s1) * (1.0f / (1.0f + expf(-s2)));
}

// ---------- batch norm over axes (node, time) ----------
__global__ void bn_reduce(const float* __restrict__ x, float* __restrict__ stats, int N, int C)
{
  const int c = blockIdx.y;
  float s = 0.f, s2 = 0.f;
  for (int n = blockIdx.x * blockDim.x + threadIdx.x; n < N; n += gridDim.x * blockDim.x) {
    const float* p = x + ((size_t)n * C + c) * 4;
    #pragma unroll
    for (int t = 0; t < 4; ++t) { float v = p[t]; s += v; s2 += v * v; }
  }
  __shared__ float ls[256], lq[256];
  const int tid = threadIdx.x;
  ls[tid] = s; lq[tid] = s2; __syncthreads();
  for (int st = 128; st > 0; st >>= 1) {
    if (tid < st) { ls[tid] += ls[tid + st]; lq[tid] += lq[tid + st]; }
    __syncthreads();
  }
  if (tid == 0) { atomicAdd(&stats[c * 2], ls[0]); atomicAdd(&stats[c * 2 + 1], lq[0]); }
}

__global__ void bn_apply(const float* __restrict__ x, const float* __restrict__ stats,
                         const float* __restrict__ g, const float* __restrict__ b,
                         const float* __restrict__ res, float* __restrict__ out,
                         int N, int C, int do_relu, int add_res)
{
  int idx = blockIdx.x * blockDim.x + threadIdx.x;      // over N*C*4
  if (idx >= N * C * 4) return;
  const int c = (idx >> 2) % C;
  const float cnt = (float)N * 4.0f;
  const float m = stats[c * 2] / cnt;
  const float var = stats[c * 2 + 1] / cnt - m * m;
  const float inv = rsqrtf(var + 1e-5f);
  float y = (x[idx] - m) * inv * g[c] + b[c];
  if (add_res) y += res[idx];
  if (do_relu) y = fmaxf(y, 0.0f);
  out[idx] = y;
}

// ---------- GAT pieces ----------
__global__ void att_scores(const float* __restrict__ hW,
                           const float* __restrict__ att_src, const float* __restrict__ att_dst,
                           float* __restrict__ aS, float* __restrict__ aD, int N, int H, int F)
{
  int idx = blockIdx.x * blockDim.x + threadIdx.x;      // over N*H
  if (idx >= N * H) return;
  const int h = idx % H, n = idx / H;
  const float* hv = hW + ((size_t)n * H + h) * F;
  float s = 0.f, d = 0.f;
  for (int f = 0; f < F; ++f) {
    const float v = hv[f];
    s += v * att_src[h * F + f];
    d += v * att_dst[h * F + f];
  }
  aS[idx] = s; aD[idx] = d;
}

__global__ void edge_max(const int* __restrict__ ei, const float* __restrict__ aS,
                         const float* __restrict__ aD, unsigned* __restrict__ emax, int E, int H)
{
  int idx = blockIdx.x * blockDim.x + threadIdx.x;      // over E*H
  if (idx >= E * H) return;
  const int h = idx % H, e = idx / H;
  const int s = ei[e], d = ei[E + e];
  float x = aS[s * H + h] + aD[d * H + h];
  x = x > 0.f ? x : 0.2f * x;                            // leaky_relu 0.2
  atomicMax(&emax[d * H + h], fenc(x));
}

__global__ void edge_exp(const int* __restrict__ ei, const float* __restrict__ aS,
                         const float* __restrict__ aD, const unsigned* __restrict__ emax,
                         float* __restrict__ ee, float* __restrict__ den, int E, int H)
{
  int idx = blockIdx.x * blockDim.x + threadIdx.x;      // over E*H
  if (idx >= E * H) return;
  const int h = idx % H, e = idx / H;
  const int s = ei[e], d = ei[E + e];
  float x = aS[s * H + h] + aD[d * H + h];
  x = x > 0.f ? x : 0.2f * x;
  const float v = expf(x - fdec(emax[d * H + h]));
  ee[idx] = v;
  atomicAdd(&den[d * H + h], v);
}

// one wave (32 lanes) per edge; 256 features -> 8 atomics/lane, all L2-resident
__global__ void edge_scatter(const int* __restrict__ ei, const float* __restrict__ hW,
                             const float* __restrict__ ee, const float* __restrict__ den,
                             float* __restrict__ out, int E, int H, int F)
{
  const int gid = blockIdx.x * blockDim.x + threadIdx.x;
  const int e = gid >> 5;
  const int lane = gid & 31;
  if (e >= E) return;
  const int s = ei[e], d = ei[E + e];
  const float a0 = ee[e * H + 0] / (den[d * H + 0] + 1e-16f);
  const float a1 = ee[e * H + 1] / (den[d * H + 1] + 1e-16f);
  const int HF = H * F;                                  // 256
  const float* hs = hW + (size_t)s * HF;
  float* od = out + (size_t)d * HF;
  for (int j = lane; j < HF; j += 32) {
    const float al = (j < F) ? a0 : a1;
    atomicAdd(&od[j], hs[j] * al);
  }
}

// bias add (+ head mean for non-concat); writes contiguous (N, cf*4) layout
__global__ void gat_final(const float* __restrict__ g, const float* __restrict__ bias,
                          float* __restrict__ dst, int N, int HF, int F, int concat)
{
  int idx = blockIdx.x * blockDim.x + threadIdx.x;
  if (concat) {
    if (idx >= N * HF) return;
    const int j = idx % HF;
    dst[idx] = g[idx] + bias[j];
  } else {
    if (idx >= N * F) return;
    const int n = idx / F, j = idx % F;
    dst[(size_t)n * F + j] = 0.5f * (g[(size_t)n * HF + j] + g[(size_t)n * HF + F + j]) + bias[j];
  }
}

__global__ void head_final(const float* __restrict__ z1, const float* __restrict__ w2,
                           const float* __restrict__ b2, float* __restrict__ out, int N, int C)
{
  int n = blockIdx.x * blockDim.x + threadIdx.x;
  if (n >= N) return;
  float s = b2[0];
  for (int c = 0; c < C; ++c) s += z1[(size_t)n * C + c] * w2[c];
  out[n] = s;
}

// ---------- host driver ----------
static inline int cdiv_i(int a, int b) { return (a + b - 1) / b; }
static inline int gemm_blocks(int M, int Nc) {
  const int waves = cdiv_i(M, 16) * (Nc / 16);
  return cdiv_i(waves * 32, 128);
}

extern "C" void kernel_launch(void* const* d_in, const int* in_sizes, int n_in,
                              void* d_out, int out_size, void* d_ws, size_t ws_size,
                              hipStream_t stream)
{
  (void)out_size; (void)ws_size;
  const int N = in_sizes[0] / (16 * 4);                  // 10000
  const int E = in_sizes[1] / 2;                         // 320000
  const float* X  = (const float*)d_in[0];
  const int*   ei = (const int*)d_in[1];

  const float* P[64];
  for (int i = 2; i < n_in && i < 64; ++i) P[i] = (const float*)d_in[i];

  const float *in_w, *in_b, *out_w1, *out_b1, *out_w2, *out_b2;
  struct Layer {
    const float *res_w, *res_b, *w1, *b1, *w2, *b2, *bn1_g, *bn1_b,
                *gat_w, *att_src, *att_dst, *gat_b, *bn2_g, *bn2_b;
  } L[2];

  if (in_sizes[2] == 512) {
    // params flattened in dict-insertion order
    int i = 2;
    in_w = P[i++]; in_b = P[i++];
    for (int l = 0; l < 2; ++l) {
      L[l].res_w = P[i++]; L[l].res_b = P[i++];
      L[l].w1 = P[i++];    L[l].b1 = P[i++];
      L[l].w2 = P[i++];    L[l].b2 = P[i++];
      L[l].bn1_g = P[i++]; L[l].bn1_b = P[i++];
      L[l].gat_w = P[i++]; L[l].att_src = P[i++]; L[l].att_dst = P[i++]; L[l].gat_b = P[i++];
      L[l].bn2_g = P[i++]; L[l].bn2_b = P[i++];
    }
    out_w1 = P[i++]; out_b1 = P[i++]; out_w2 = P[i++]; out_b2 = P[i++];
  } else {
    // JAX pytree order: dict keys sorted alphabetically
    int i = 2;
    in_b = P[i++]; in_w = P[i++];
    for (int l = 0; l < 2; ++l) {
      L[l].att_dst = P[i++]; L[l].att_src = P[i++];
      L[l].b1 = P[i++];      L[l].b2 = P[i++];
      L[l].bn1_b = P[i++];   L[l].bn1_g = P[i++];
      L[l].bn2_b = P[i++];   L[l].bn2_g = P[i++];
      L[l].gat_b = P[i++];   L[l].gat_w = P[i++];
      L[l].res_b = P[i++];   L[l].res_w = P[i++];
      L[l].w1 = P[i++];      L[l].w2 = P[i++];
    }
    out_b1 = P[i++]; out_b2 = P[i++]; out_w1 = P[i++]; out_w2 = P[i++];
  }

  // workspace carve (~59 MB)
  char* base = (char*)d_ws;
  size_t off = 0;
  auto carve = [&](size_t bytes) -> void* {
    off = (off + 255) & ~(size_t)255;
    void* p = base + off; off += bytes; return p;
  };
  float*    bufA  = (float*)carve((size_t)N * 64 * 4 * sizeof(float));   // x ping
  float*    bufB  = (float*)carve((size_t)N * 64 * 4 * sizeof(float));   // x pong
  float*    bufR  = (float*)carve((size_t)N * 64 * 4 * sizeof(float));   // residual
  float*    bufH  = (float*)carve((size_t)N * 128 * sizeof(float));      // gated conv / z1
  float*    bufHW = (float*)carve((size_t)N * 256 * sizeof(float));      // h @ gat_w
  float*    gout  = (float*)carve((size_t)N * 256 * sizeof(float));      // GAT scatter accumulator
  float*    aS    = (float*)carve((size_t)N * 2 * sizeof(float));
  float*    aD    = (float*)carve((size_t)N * 2 * sizeof(float));
  unsigned* emax  = (unsigned*)carve((size_t)N * 2 * sizeof(unsigned));
  float*    den   = (float*)carve((size_t)N * 2 * sizeof(float));
  float*    ee    = (float*)carve((size_t)E * 2 * sizeof(float));
  float*    stats = (float*)carve(128 * sizeof(float));

  // ---- input 1x1 conv: (N*4,16) x (16,32) -> NCT(32) ----
  gemm_wmma_f32<1, 1, 0><<<gemm_blocks(4 * N, 32), 128, 0, stream>>>(
      X, in_w, in_b, bufA, 4 * N, 32, 16, /*lda*/ 64, /*ldbk*/ 1, /*ldbj*/ 16, /*ldc*/ 128);

  const int DIL[2] = {1, 2};
  float* x = bufA;
  int cin = 32;
  for (int l = 0; l < 2; ++l) {
    const int concat = (l == 0) ? 1 : 0;
    const int cout = 32;
    const int cf = concat ? 64 : 32;
    float* xn = (l == 0) ? bufB : bufA;

    // residual 1x1 conv: (N*4,cin) x (cin,cf) -> NCT(cf)
    gemm_wmma_f32<1, 1, 0><<<gemm_blocks(4 * N, cf), 128, 0, stream>>>(
        x, L[l].res_w, L[l].res_b, bufR, 4 * N, cf, cin, cin * 4, 1, cin, cf * 4);

    // gated dilated conv -> bufH (N,32,4)
    dconv_gate<<<cdiv_i(N * cout * 4, 256), 256, 0, stream>>>(
        x, L[l].w1, L[l].b1, L[l].w2, L[l].b2, bufH, N, cin, cout, DIL[l]);

    // bn1 (in place)
    zero_u32<<<1, 128, 0, stream>>>((unsigned*)stats, 128);
    { dim3 g(40, cout); bn_reduce<<<g, 256, 0, stream>>>(bufH, stats, N, cout); }
    bn_apply<<<cdiv_i(N * cout * 4, 256), 256, 0, stream>>>(
        bufH, stats, L[l].bn1_g, L[l].bn1_b, nullptr, bufH, N, cout, 0, 0);

    // GAT projection: (N,128) x (128,256) -> (N,256)
    gemm_wmma_f32<0, 0, 0><<<gemm_blocks(N, 256), 128, 0, stream>>>(
        bufH, L[l].gat_w, nullptr, bufHW, N, 256, 128, 128, /*ldbk*/ 256, /*ldbj*/ 1, 256);

    // attention scores
    att_scores<<<cdiv_i(N * 2, 256), 256, 0, stream>>>(
        bufHW, L[l].att_src, L[l].att_dst, aS, aD, N, 2, 128);

    // clear scratch, then the three edge passes
    zero_u32<<<cdiv_i(N * 2, 256), 256, 0, stream>>>(emax, N * 2);
    zero_u32<<<cdiv_i(N * 2, 256), 256, 0, stream>>>((unsigned*)den, N * 2);
    zero_u32<<<cdiv_i(N * 256, 256), 256, 0, stream>>>((unsigned*)gout, N * 256);
    edge_max<<<cdiv_i(E * 2, 256), 256, 0, stream>>>(ei, aS, aD, emax, E, 2);
    edge_exp<<<cdiv_i(E * 2, 256), 256, 0, stream>>>(ei, aS, aD, emax, ee, den, E, 2);
    edge_scatter<<<cdiv_i(E * 32, 256), 256, 0, stream>>>(ei, bufHW, ee, den, gout, E, 2, 128);

    // bias (+ head-mean) -> contiguous (N, cf*4) in bufHW (hW no longer needed)
    gat_final<<<cdiv_i(N * (concat ? 256 : 128), 256), 256, 0, stream>>>(
        gout, L[l].gat_b, bufHW, N, 256, 128, concat);

    // bn2 + residual (+relu if concat) -> next x
    zero_u32<<<1, 128, 0, stream>>>((unsigned*)stats, 128);
    { dim3 g(40, cf); bn_reduce<<<g, 256, 0, stream>>>(bufHW, stats, N, cf); }
    bn_apply<<<cdiv_i(N * cf * 4, 256), 256, 0, stream>>>(
        bufHW, stats, L[l].bn2_g, L[l].bn2_b, bufR, xn, N, cf, concat, 1);

    x = xn; cin = cf;
  }

  // ---- output head: relu(x[:,:,-1] @ out_w1^T + b1) via WMMA, then final dot ----
  gemm_wmma_f32<2, 0, 1><<<gemm_blocks(N, 64), 128, 0, stream>>>(
      x, out_w1, out_b1, bufH, N, 64, 32, /*lda*/ cin * 4, 1, 32, 64);
  head_final<<<cdiv_i(N, 256), 256, 0, stream>>>(bufH, out_w2, out_b2, (float*)d_out, N, 64);
}